// BuiltCNOT_15762529977052
// MI455X (gfx1250) — compile-verified
//
#include <hip/hip_runtime.h>
#include <cstdint>

// out = state @ M where M is the dense CNOT(c=0,t=1) matrix on 13 qubits.
// M is a permutation (involution), so:
//     out[b, j] = state[b, j ^ 2048]  if (j & 4096) else state[b, j]
// => pure permuted copy: 64 MB read + 64 MB write = 128 MB of HBM traffic,
//    ~5.5 us floor at 23.3 TB/s. A WMMA formulation would be 275 GFLOP of
//    f32 work plus a 256 MB read of M -- strictly worse, so no matrix ops.
//
// Data path: CDNA5 async DMA (global -> LDS -> global), GVS addressing
// (SGPR base + 32-bit VGPR byte offset), 4 x b128 per lane issued as one
// s_clause, synchronized with s_wait_asynccnt. Each async instruction moves
// a contiguous 512 B per wave on both sides (flipped permutation bits are
// above float4 granularity).

#define BLOCK_THREADS   256
#define VEC_PER_THREAD  4              // 4 x float4 = 64 B per thread
#define VEC_PER_BLOCK   (BLOCK_THREADS * VEC_PER_THREAD)   // 1024 float4s
#define VEC_PER_ROW     2048u          // N/4 = 8192/4 float4s per row
#define CTRL_BIT_V      10             // bit 12 of column j == bit 10 of j4
#define TGT_BIT_V       9              // bit 11 of column j == bit  9 of j4

__global__ __launch_bounds__(BLOCK_THREADS)
void cnot_permute_async(const float* __restrict__ src,
                        float* __restrict__ dst)
{
    // 256 threads * 4 slots * 16 B = 16 KB LDS staging per block.
    __shared__ __align__(16) float stage[VEC_PER_BLOCK * 4];

    // Low 32 bits of a generic pointer to __shared__ == LDS byte offset.
    const uint32_t lds_base = (uint32_t)(uintptr_t)(&stage[0]);

    const uint32_t tid = threadIdx.x;
    const uint32_t bs  = blockIdx.x * (uint32_t)VEC_PER_BLOCK;

    uint32_t soff[VEC_PER_THREAD];   // source byte offset (permuted column)
    uint32_t doff[VEC_PER_THREAD];   // destination byte offset (identity)
    uint32_t loff[VEC_PER_THREAD];   // LDS byte address

#pragma unroll
    for (int k = 0; k < VEC_PER_THREAD; ++k) {
        const uint32_t vec_idx = bs + (uint32_t)k * BLOCK_THREADS + tid;
        const uint32_t j4      = vec_idx & (VEC_PER_ROW - 1u);
        const uint32_t src_j4  = j4 ^ (((j4 >> CTRL_BIT_V) & 1u) << TGT_BIT_V);
        const uint32_t src_vec = (vec_idx & ~(VEC_PER_ROW - 1u)) | src_j4;

        soff[k] = src_vec * 16u;                                  // < 64 MB, fits u32
        doff[k] = vec_idx * 16u;
        loff[k] = lds_base + ((uint32_t)k * BLOCK_THREADS + tid) * 16u;
    }

    // Phase 1: one clause of 4 async DMA loads, global -> LDS (GVS mode:
    // vdst = LDS byte addr, vaddr = 32-bit byte offset, saddr = 64-bit base).
    asm volatile(
        "s_clause 0x3\n\t"
        "global_load_async_to_lds_b128 %0, %4, %8\n\t"
        "global_load_async_to_lds_b128 %1, %5, %8\n\t"
        "global_load_async_to_lds_b128 %2, %6, %8\n\t"
        "global_load_async_to_lds_b128 %3, %7, %8"
        :: "v"(loff[0]), "v"(loff[1]), "v"(loff[2]), "v"(loff[3]),
           "v"(soff[0]), "v"(soff[1]), "v"(soff[2]), "v"(soff[3]),
           "s"(src)
        : "memory");

    // The wave's async loads must land in LDS before the DMA stores read it
    // (async LDS accesses may complete out of order).
    asm volatile("s_wait_asynccnt 0" ::: "memory");

    // Phase 2: one clause of 4 async DMA stores, LDS -> global
    // (vaddr = 32-bit byte offset, vsrc = LDS byte addr, saddr = 64-bit base).
    asm volatile(
        "s_clause 0x3\n\t"
        "global_store_async_from_lds_b128 %4, %0, %8\n\t"
        "global_store_async_from_lds_b128 %5, %1, %8\n\t"
        "global_store_async_from_lds_b128 %6, %2, %8\n\t"
        "global_store_async_from_lds_b128 %7, %3, %8"
        :: "v"(loff[0]), "v"(loff[1]), "v"(loff[2]), "v"(loff[3]),
           "v"(doff[0]), "v"(doff[1]), "v"(doff[2]), "v"(doff[3]),
           "s"(dst)
        : "memory");
    // s_endpgm performs an implicit wait-idle, draining outstanding ASYNCcnt.
}

extern "C" void kernel_launch(void* const* d_in, const int* in_sizes, int n_in,
                              void* d_out, int out_size, void* d_ws, size_t ws_size,
                              hipStream_t stream)
{
    const float* state = (const float*)d_in[0];   // [2048, 8192] f32
    // d_in[1] (the dense CNOT matrix M) is deliberately unused: the matmul is
    // a column permutation; touching M's 256 MB would only burn bandwidth.
    float* out = (float*)d_out;

    const uint32_t total_vec = (uint32_t)(out_size / 4);   // 4,194,304 float4s
    // 2048*8192 floats -> total_vec divides VEC_PER_BLOCK (1024) exactly,
    // so every block is full and the kernel needs no bounds guards.
    const uint32_t grid = total_vec / VEC_PER_BLOCK;       // 4096 blocks

    cnot_permute_async<<<grid, BLOCK_THREADS, 0, stream>>>(state, out);
}